// FFTConv_37374805409874
// MI455X (gfx1250) — compile-verified
//
#include <hip/hip_runtime.h>
#include <math.h>

// FFT convolution (Hyena-style long depthwise conv):
//   B=2, L=8192, D=1024, filter_len=8192, fft_size=16384
// One workgroup per (b, d) column; transform fully LDS-resident (256 KB + 2 KB
// twiddle tables < 320 KB/WGP). Radix-16 Stockham stages run on the f32 tensor
// path (V_WMMA_F32_16X16X4_F32 chained x4 for K=16, x4 for complex). Column
// loads AND the result store are DMA'd by the Tensor Data Mover
// (tensor_load_to_lds / tensor_store_from_lds, TENSORcnt).

typedef __attribute__((ext_vector_type(2))) float v2f;
typedef __attribute__((ext_vector_type(8))) float v8f;
typedef __attribute__((ext_vector_type(4))) unsigned int v4u;
typedef __attribute__((ext_vector_type(4))) int v4i;
typedef __attribute__((ext_vector_type(8))) int v8i;

#define FFT_N   16384
#define LSEQ    8192
#define DCH     1024
#define TPB     256
#define NWAVES  8
#define TWO_PI  6.283185307179586f

#define HAVE_TDM_LOAD  (__has_builtin(__builtin_amdgcn_tensor_load_to_lds) && \
                        __has_builtin(__builtin_amdgcn_s_wait_tensorcnt))
#define HAVE_TDM_STORE (__has_builtin(__builtin_amdgcn_tensor_store_from_lds) && \
                        __has_builtin(__builtin_amdgcn_s_wait_tensorcnt))

#if HAVE_TDM_LOAD || HAVE_TDM_STORE
// ---------------------------------------------------------------------------
// D# for a strided f32 column: 2D tensor, data_size=4B, tile 1 x LSEQ (one
// f32 per row, row stride = DCH elements). Bit packing per ISA
// CDNA5 ISA. All fields wave-uniform; tensor ops ignore EXEC,
// so a wave-uniform branch issues each DMA exactly once.
// ---------------------------------------------------------------------------
struct TdmDesc {
  v4u g0;
  v8i g1;
  v4i g2;
  v4i g3;
};

__device__ __forceinline__ TdmDesc tdm_column_desc(const void* gaddr,
                                                   unsigned lds_off) {
  const unsigned long long ga = (unsigned long long)(uintptr_t)gaddr;
  TdmDesc t;
  t.g0[0] = 1u;                                   // count=1, user, no gather
  t.g0[1] = lds_off;                              // lds_addr (bytes)
  t.g0[2] = (unsigned)(ga & 0xFFFFFFFFu);         // global_addr[31:0]
  t.g0[3] = (unsigned)((ga >> 32) & 0x1FFFFFFu)   // global_addr[56:32]
            | (2u << 30);                         // type=2 ("image")
  t.g1[0] = 0x00020000;        // wg_mask=0, data_size=2 (4B), no flags
  t.g1[1] = (int)(DCH << 16);  // tensor_dim0[15:0]=1024 at bits [63:48]
  t.g1[2] = (int)(LSEQ << 16); // tensor_dim1[15:0]=8192 at bits [111:80]
  t.g1[3] = 0x00010000;        // tile_dim0=1 at bits [127:112]
  t.g1[4] = LSEQ;              // tile_dim1=8192 at [143:128]; tile_dim2=0
  t.g1[5] = DCH;               // tensor_dim0_stride low32 = 1024
  t.g1[6] = 0;                 // stride0 hi=0; tensor_dim1_stride[15:0]=0
  t.g1[7] = (int)((LSEQ * DCH) >> 16); // tensor_dim1_stride[47:16] (unused)
  t.g2 = (v4i){1, 0, 0, 0};    // tensor_dim2=1; tile_dim3=0
  t.g3 = (v4i){0, 0, 0, 0};
  return t;
}
#endif

#if HAVE_TDM_LOAD
__device__ __forceinline__ void tdm_load_column(const float* gsrc,
                                                unsigned lds_off) {
  TdmDesc t = tdm_column_desc((const void*)gsrc, lds_off);
#if defined(__clang_major__) && (__clang_major__ >= 23)
  v8i g4 = {0, 0, 0, 0, 0, 0, 0, 0};
  __builtin_amdgcn_tensor_load_to_lds(t.g0, t.g1, t.g2, t.g3, g4, 0);
#else
  __builtin_amdgcn_tensor_load_to_lds(t.g0, t.g1, t.g2, t.g3, 0);
#endif
}
#endif

#if HAVE_TDM_STORE
__device__ __forceinline__ void tdm_store_column(float* gdst,
                                                 unsigned lds_off) {
  TdmDesc t = tdm_column_desc((const void*)gdst, lds_off);
#if defined(__clang_major__) && (__clang_major__ >= 23)
  v8i g4 = {0, 0, 0, 0, 0, 0, 0, 0};
  __builtin_amdgcn_tensor_store_from_lds(t.g0, t.g1, t.g2, t.g3, g4, 0);
#else
  __builtin_amdgcn_tensor_store_from_lds(t.g0, t.g1, t.g2, t.g3, 0);
#endif
}
#endif

// ---------------------------------------------------------------------------
// Radix-16 Stockham stage, 16 sub-DFTs per wave batch via WMMA.
//   out[q + s*(16p + u)] = ( sum_c W16[u][c] * in[q + s*(p + m*c)] ) * Wn^{u*p}
// Twiddles via two-level LDS table: W_N^E = TwHi[E>>7] * TwLo[E&127],
// E = ((u*p) mod n) * (N/n); tables store forward sign (e^{-i*theta}).
// A-fragment (ISA 7.12.2, 32-bit A 16x4): lane = M + 16*(Krel>=2), vreg=Krel&1.
// B (4x16) mirrored; C/D (16x16 f32): lane L, vreg v -> (M=v+8*(L>=16), N=L&15)
// ---------------------------------------------------------------------------
__device__ __forceinline__ void radix16_wmma_stage(
    const float* __restrict__ Xre, const float* __restrict__ Xim,
    float* __restrict__ Yre, float* __restrict__ Yim,
    const float2* __restrict__ TwHi, const float2* __restrict__ TwLo,
    int n, int logs, int logmult, bool fwd, int lane, int wave)
{
  const int m  = n >> 4;
  const int M  = lane & 15;          // A row / B column / D column index
  const int hi = (lane >> 4) & 1;    // lane half selects Krel/M-high

  // DFT-16 matrix fragments (A operand): W16^j = TwHi[8j].
  v2f Wr[4], Wi[4], nWi[4];
#pragma unroll
  for (int kb = 0; kb < 4; ++kb) {
#pragma unroll
    for (int r = 0; r < 2; ++r) {
      const int K = 4 * kb + 2 * hi + r;
      const float2 w = TwHi[((M * K) & 15) << 3];
      const float wr = w.x;
      const float wi = fwd ? w.y : -w.y;
      Wr[kb][r] = wr;
      Wi[kb][r] = wi;
      nWi[kb][r] = -wi;
    }
  }

  const int ngroups = (m << logs) >> 4;   // 64 for every radix-16 stage here
  for (int g = wave; g < ngroups; g += NWAVES) {
    const int t = (g << 4) + M;           // this lane's column -> instance id
    const int p = t >> logs;
    const int q = t - (p << logs);

    // Gather B fragments (complex data), row c = K of the 16-point DFT.
    v2f Br[4], Bi[4];
#pragma unroll
    for (int kb = 0; kb < 4; ++kb) {
#pragma unroll
      for (int r = 0; r < 2; ++r) {
        const int c   = 4 * kb + 2 * hi + r;
        const int idx = q + ((p + m * c) << logs);
        Br[kb][r] = Xre[idx];
        Bi[kb][r] = Xim[idx];
      }
    }

    // Complex matmul: Cre = Wr*Br - Wi*Bi ; Cim = Wr*Bi + Wi*Br
    v8f Cre = {0.f, 0.f, 0.f, 0.f, 0.f, 0.f, 0.f, 0.f};
    v8f Cim = {0.f, 0.f, 0.f, 0.f, 0.f, 0.f, 0.f, 0.f};
#pragma unroll
    for (int kb = 0; kb < 4; ++kb) {
      Cre = __builtin_amdgcn_wmma_f32_16x16x4_f32(false, Wr[kb],  false, Br[kb], (short)0, Cre, false, false);
      Cre = __builtin_amdgcn_wmma_f32_16x16x4_f32(false, nWi[kb], false, Bi[kb], (short)0, Cre, false, false);
      Cim = __builtin_amdgcn_wmma_f32_16x16x4_f32(false, Wr[kb],  false, Bi[kb], (short)0, Cim, false, false);
      Cim = __builtin_amdgcn_wmma_f32_16x16x4_f32(false, Wi[kb],  false, Br[kb], (short)0, Cim, false, false);
    }

    // Twiddle (table product) + scatter. D element (u, j=M): u = v + 8*hi.
#pragma unroll
    for (int v = 0; v < 8; ++v) {
      const int u = v + 8 * hi;
      const int E = ((u * p) & (n - 1)) << logmult;
      const float2 th = TwHi[E >> 7];
      const float2 tl = TwLo[E & 127];
      const float cv  = th.x * tl.x - th.y * tl.y;
      const float svf = th.x * tl.y + th.y * tl.x;
      const float sv  = fwd ? svf : -svf;
      const float yr = Cre[v] * cv - Cim[v] * sv;
      const float yi = Cre[v] * sv + Cim[v] * cv;
      const int dst = q + (((p << 4) + u) << logs);
      Yre[dst] = yr;
      Yim[dst] = yi;
    }
  }
}

// Final radix-4 stage: n=4, s=4096, m=1 (p=0 => all twiddles are 1).
__device__ __forceinline__ void radix4_stage(
    const float* __restrict__ Xre, const float* __restrict__ Xim,
    float* __restrict__ Yre, float* __restrict__ Yim, bool fwd, int tid)
{
  const float sgn = fwd ? -1.0f : 1.0f;
  const int s = FFT_N >> 2;            // 4096
  for (int q = tid; q < s; q += TPB) {
    const float ar = Xre[q],         ai = Xim[q];
    const float br = Xre[q + s],     bi = Xim[q + s];
    const float cr = Xre[q + 2 * s], ci = Xim[q + 2 * s];
    const float dr = Xre[q + 3 * s], di = Xim[q + 3 * s];
    const float t0r = ar + cr, t0i = ai + ci;
    const float t1r = ar - cr, t1i = ai - ci;
    const float t2r = br + dr, t2i = bi + di;
    const float t3r = br - dr, t3i = bi - di;
    const float u3r = -sgn * t3i;      // sgn*i*(t3)
    const float u3i =  sgn * t3r;
    Yre[q]         = t0r + t2r;  Yim[q]         = t0i + t2i;
    Yre[q + s]     = t1r + u3r;  Yim[q + s]     = t1i + u3i;
    Yre[q + 2 * s] = t0r - t2r;  Yim[q + 2 * s] = t0i - t2i;
    Yre[q + 3 * s] = t1r - u3r;  Yim[q + 3 * s] = t1i - u3i;
  }
}

// Full 16384-point complex FFT: radices 16,16,16,4. Ping-pong S->T->S->T->S.
// Result lands back in (Sre, Sim). fwd=true forward, false inverse (unscaled).
__device__ __forceinline__ void fft16384(
    float* Sre, float* Sim, float* Tre, float* Tim,
    const float2* TwHi, const float2* TwLo,
    bool fwd, int tid, int lane, int wave)
{
  radix16_wmma_stage(Sre, Sim, Tre, Tim, TwHi, TwLo, 16384, 0, 0, fwd, lane, wave);
  __syncthreads();
  radix16_wmma_stage(Tre, Tim, Sre, Sim, TwHi, TwLo, 1024, 4, 4, fwd, lane, wave);
  __syncthreads();
  radix16_wmma_stage(Sre, Sim, Tre, Tim, TwHi, TwLo, 64, 8, 8, fwd, lane, wave);
  __syncthreads();
  radix4_stage(Tre, Tim, Sre, Sim, fwd, tid);
  __syncthreads();
}

extern __shared__ float smem[];

__global__ void __launch_bounds__(TPB)
fftconv_kernel(const float* __restrict__ x, const float* __restrict__ h,
               const float* __restrict__ bias, float* __restrict__ out)
{
  float* Are = smem;
  float* Aim = smem + FFT_N;
  float* Bre = smem + 2 * FFT_N;
  float* Bim = smem + 3 * FFT_N;
  float2* TwHi = (float2*)(smem + 4 * FFT_N);   // 128 x e^{-2pi i k/128}
  float2* TwLo = TwHi + 128;                    // 128 x e^{-2pi i j/16384}

  const int tid  = threadIdx.x;
  const int lane = tid & 31;
  const int wave = tid >> 5;
  const int b    = blockIdx.x / DCH;
  const int d    = blockIdx.x - b * DCH;

  const float* xcol = x + (size_t)b * LSEQ * DCH + d;
  const float* hcol = h + d;

#if HAVE_TDM_LOAD
  // DMA both strided columns into LDS (re <- x, im <- h); wave 0 issues.
  if (wave == 0) {
    tdm_load_column(xcol, (unsigned)(uintptr_t)(void*)Are);
    tdm_load_column(hcol, (unsigned)(uintptr_t)(void*)Aim);
  }
  // Meanwhile: zero-pad the upper half and build twiddle tables.
  for (int l = LSEQ + tid; l < FFT_N; l += TPB) {
    Are[l] = 0.f;
    Aim[l] = 0.f;
  }
  {
    float sv, cv;
    if (tid < 128) {
      sincosf(-(TWO_PI / 128.0f) * (float)tid, &sv, &cv);
      TwHi[tid] = make_float2(cv, sv);
    } else {
      const int j = tid - 128;
      sincosf(-(TWO_PI / (float)FFT_N) * (float)j, &sv, &cv);
      TwLo[j] = make_float2(cv, sv);
    }
  }
  if (wave == 0) __builtin_amdgcn_s_wait_tensorcnt(0);
  __syncthreads();
#else
  for (int l = tid; l < FFT_N; l += TPB) {
    float re = 0.f, im = 0.f;
    if (l < LSEQ) {
      re = xcol[(size_t)l * DCH];
      im = hcol[(size_t)l * DCH];
    }
    Are[l] = re;
    Aim[l] = im;
  }
  {
    float sv, cv;
    if (tid < 128) {
      sincosf(-(TWO_PI / 128.0f) * (float)tid, &sv, &cv);
      TwHi[tid] = make_float2(cv, sv);
    } else {
      const int j = tid - 128;
      sincosf(-(TWO_PI / (float)FFT_N) * (float)j, &sv, &cv);
      TwLo[j] = make_float2(cv, sv);
    }
  }
  __syncthreads();
#endif

  // Forward FFT of packed z = x + i*h. Result in A.
  fft16384(Are, Aim, Bre, Bim, TwHi, TwLo, true, tid, lane, wave);

  // Unpack X_f, H_f from Z and multiply:  X_f = (Z[k]+conj(Z[N-k]))/2,
  // H_f = -i*(Z[k]-conj(Z[N-k]))/2,  Y = X_f*H_f / N  (Hermitian).
  const float invN = 1.0f / (float)FFT_N;
  for (int k = tid; k <= FFT_N / 2; k += TPB) {
    const int nk = (FFT_N - k) & (FFT_N - 1);
    const float zr = Are[k],  zi = Aim[k];
    const float cr = Are[nk], ci = Aim[nk];
    const float afr = 0.5f * (zr + cr);
    const float afi = 0.5f * (zi - ci);
    const float bfr = 0.5f * (zi + ci);
    const float bfi = -0.5f * (zr - cr);
    const float yr = (afr * bfr - afi * bfi) * invN;
    const float yi = (afr * bfi + afi * bfr) * invN;
    Bre[k]  = yr;  Bim[k]  = yi;
    Bre[nk] = yr;  Bim[nk] = -yi;   // conjugate-symmetric spectrum
  }
  __syncthreads();

  // Inverse FFT (conjugate twiddles; 1/N already folded in). Result in B.
  fft16384(Bre, Bim, Are, Aim, TwHi, TwLo, false, tid, lane, wave);

  // y[t] = Re(inv)[t] + bias[d] for t < LSEQ.
  const float bval = bias[d];
  float* ocol = out + (size_t)b * LSEQ * DCH + d;
#if HAVE_TDM_STORE
  // Add bias in LDS, then DMA the strided column back out with the TDM.
  for (int t = tid; t < LSEQ; t += TPB) {
    Bre[t] += bval;
  }
  __syncthreads();
  if (wave == 0) {
    tdm_store_column(ocol, (unsigned)(uintptr_t)(void*)Bre);
    __builtin_amdgcn_s_wait_tensorcnt(0);
  }
#else
  for (int t = tid; t < LSEQ; t += TPB) {
    ocol[(size_t)t * DCH] = Bre[t] + bval;
  }
#endif
}

extern "C" void kernel_launch(void* const* d_in, const int* in_sizes, int n_in,
                              void* d_out, int out_size, void* d_ws, size_t ws_size,
                              hipStream_t stream) {
  (void)in_sizes; (void)n_in; (void)d_ws; (void)ws_size; (void)out_size;
  const float* x    = (const float*)d_in[0];
  const float* h    = (const float*)d_in[1];
  const float* bias = (const float*)d_in[2];
  float* out        = (float*)d_out;

  const dim3 grid(2 * DCH);   // one workgroup per (b, d) column
  const dim3 block(TPB);
  const size_t shbytes = (4u * (size_t)FFT_N + 2u * 128u * 2u) * sizeof(float);
  hipLaunchKernelGGL(fftconv_kernel, grid, block, shbytes, stream,
                     x, h, bias, out);
}